// GNN_13469017440589
// MI455X (gfx1250) — compile-verified
//
#include <hip/hip_runtime.h>
#include <math.h>

typedef __attribute__((ext_vector_type(2))) float v2f;
typedef __attribute__((ext_vector_type(8))) float v8f;

// ---------------------------------------------------------------------------
// K1: zero the degree buffer
// ---------------------------------------------------------------------------
__global__ void k_zero(float* __restrict__ deg, int n) {
  int i = blockIdx.x * blockDim.x + threadIdx.x;
  if (i < n) deg[i] = 0.0f;
}

// ---------------------------------------------------------------------------
// K2: in-degree count (self-loop added analytically later as +1)
// ---------------------------------------------------------------------------
__global__ void k_degree(const int* __restrict__ dst, float* __restrict__ deg,
                         int nE) {
  int e = blockIdx.x * blockDim.x + threadIdx.x;
  if (e < nE) atomicAdd(&deg[dst[e]], 1.0f);
}

// ---------------------------------------------------------------------------
// K3: per 16-node tile (one wave32 each):
//   dinv[i] = rsqrt(deg[i] + 1)            (lanes 0..15)
//   xw      = x[i,0:2] @ W1[2,8]           (V_WMMA_F32_16X16X4_F32, exact f32)
//   acc1[i] = xw * dinv[i]^2               (self-loop contribution seed)
// A 16x4 layout: lanes 0-15 = (K0,K1) of row=lane; lanes 16-31 = (K2,K3)=pad0.
// B 4x16 layout: lanes 0-15 = (K0,K1) of col=lane; cols>=8 and K>=2 are pad0.
// C 16x16: lane holds col N=lane%16, VGPR r holds row M=r+(lane<16?0:8).
// ---------------------------------------------------------------------------
__global__ void k_node1(const float* __restrict__ x, const float* __restrict__ W1,
                        float* __restrict__ dinv, float* __restrict__ acc1,
                        int n) {
  int lane = threadIdx.x & 31;
  int wave = (blockIdx.x * blockDim.x + threadIdx.x) >> 5;
  int base = wave * 16;
  if (base >= n) return;  // wave-uniform: n % 16 == 0

  // finalize dinv for this tile's 16 nodes (lanes 0..15)
  float di = 0.0f;
  if (lane < 16) {
    float cnt = dinv[base + lane];
    di = rsqrtf(cnt + 1.0f);  // deg incl. self-loop, always > 0
    dinv[base + lane] = di;
  }

  // A tile: x rows, K-padded to 4
  v2f a = {0.0f, 0.0f};
  if (lane < 16) {
    const float* xp = x + (size_t)(base + lane) * 2;
    a.x = xp[0];
    a.y = xp[1];
  }
  // B tile: W1 [2][8] row-major, padded to 4x16
  v2f b = {0.0f, 0.0f};
  if (lane < 8) {
    b.x = W1[lane];      // K=0 row
    b.y = W1[8 + lane];  // K=1 row
  }

  v8f c = {};
  c = __builtin_amdgcn_wmma_f32_16x16x4_f32(false, a, false, b,
                                            (short)0, c, false, false);

  int ncol = lane & 15;
  int moff = (lane < 16) ? 0 : 8;
  if (ncol < 8) {
#pragma unroll
    for (int r = 0; r < 8; ++r) {
      int m = r + moff;
      float dm = __shfl(di, m, 32);  // dinv of row m's node
      acc1[(size_t)(base + m) * 8 + ncol] = c[r] * dm * dm;
    }
  }
}

// ---------------------------------------------------------------------------
// K4: layer-1 edge pass. Gather x[src] (8B), apply W1 on the fly (W1 is
// uniform -> scalar loads), scatter-add 8 floats into acc1[dst].
// ---------------------------------------------------------------------------
__global__ void k_edge1(const int* __restrict__ src, const int* __restrict__ dst,
                        const float* __restrict__ x, const float* __restrict__ W1,
                        const float* __restrict__ dinv, float* __restrict__ acc1,
                        int nE) {
  int e = blockIdx.x * blockDim.x + threadIdx.x;
  if (e >= nE) return;
  int s = src[e];
  int d = dst[e];
  float norm = dinv[s] * dinv[d];
  const float* xp = x + (size_t)s * 2;
  float x0 = xp[0] * norm;
  float x1 = xp[1] * norm;
  float* out = acc1 + (size_t)d * 8;
#pragma unroll
  for (int j = 0; j < 8; ++j) {
    atomicAdd(&out[j], fmaf(x0, W1[j], x1 * W1[8 + j]));
  }
}

// ---------------------------------------------------------------------------
// K5: per 16-node tile (one wave32):
//   h  = relu(acc1 + b1)                 (applied elementwise during A load)
//   hw = h @ W2[8,2]                     (two chained V_WMMA_F32_16X16X4_F32,
//                                         K=8 split as 4+4 via C accumulation)
//   d_out[i] = b2 + hw[i]*dinv[i]^2      (bias + self-loop seed)
// ---------------------------------------------------------------------------
__global__ void k_node2(const float* __restrict__ acc1, const float* __restrict__ b1,
                        const float* __restrict__ W2, const float* __restrict__ b2,
                        const float* __restrict__ dinv, float* __restrict__ hw,
                        float* __restrict__ out, int n) {
  int lane = threadIdx.x & 31;
  int wave = (blockIdx.x * blockDim.x + threadIdx.x) >> 5;
  int base = wave * 16;
  if (base >= n) return;  // wave-uniform

  int row = lane & 15;
  int koff = (lane < 16) ? 0 : 2;  // K pair this lane supplies
  const float* ap = acc1 + (size_t)(base + row) * 8;

  // A tiles with fused bias+ReLU: wmma0 covers K=0..3, wmma1 covers K=4..7
  v2f a0, a1;
  a0.x = fmaxf(ap[koff + 0] + b1[koff + 0], 0.0f);
  a0.y = fmaxf(ap[koff + 1] + b1[koff + 1], 0.0f);
  a1.x = fmaxf(ap[koff + 4] + b1[koff + 4], 0.0f);
  a1.y = fmaxf(ap[koff + 5] + b1[koff + 5], 0.0f);

  // B tiles: W2 [8][2] row-major -> 4x16 slabs, cols >= 2 are zero
  int ncol = lane & 15;
  v2f b0 = {0.0f, 0.0f};
  v2f b1v = {0.0f, 0.0f};
  if (ncol < 2) {
    b0.x  = W2[(koff + 0) * 2 + ncol];
    b0.y  = W2[(koff + 1) * 2 + ncol];
    b1v.x = W2[(koff + 4) * 2 + ncol];
    b1v.y = W2[(koff + 5) * 2 + ncol];
  }

  v8f c = {};
  c = __builtin_amdgcn_wmma_f32_16x16x4_f32(false, a0, false, b0,
                                            (short)0, c, false, false);
  c = __builtin_amdgcn_wmma_f32_16x16x4_f32(false, a1, false, b1v,
                                            (short)0, c, false, false);

  int moff = (lane < 16) ? 0 : 8;
  if (ncol < 2) {
    float bias = b2[ncol];
#pragma unroll
    for (int r = 0; r < 8; ++r) {
      int node = base + r + moff;
      float di = dinv[node];
      float v = c[r];
      hw[(size_t)node * 2 + ncol] = v;
      out[(size_t)node * 2 + ncol] = bias + v * di * di;
    }
  }
}

// ---------------------------------------------------------------------------
// K6: layer-2 edge pass. Gather hw[src] (8B), scatter-add 2 floats to d_out.
// ---------------------------------------------------------------------------
__global__ void k_edge2(const int* __restrict__ src, const int* __restrict__ dst,
                        const float* __restrict__ hw, const float* __restrict__ dinv,
                        float* __restrict__ out, int nE) {
  int e = blockIdx.x * blockDim.x + threadIdx.x;
  if (e >= nE) return;
  int s = src[e];
  int d = dst[e];
  float norm = dinv[s] * dinv[d];
  const float* hp = hw + (size_t)s * 2;
  atomicAdd(&out[(size_t)d * 2 + 0], hp[0] * norm);
  atomicAdd(&out[(size_t)d * 2 + 1], hp[1] * norm);
}

// ---------------------------------------------------------------------------
extern "C" void kernel_launch(void* const* d_in, const int* in_sizes, int n_in,
                              void* d_out, int out_size, void* d_ws, size_t ws_size,
                              hipStream_t stream) {
  const float* x  = (const float*)d_in[0];
  const int* edge = (const int*)d_in[1];
  const float* W1 = (const float*)d_in[2];
  const float* b1 = (const float*)d_in[3];
  const float* W2 = (const float*)d_in[4];
  const float* b2 = (const float*)d_in[5];

  int nN = in_sizes[0] / 2;   // 1,000,000
  int nE = in_sizes[1] / 2;   // 10,000,000
  const int* src = edge;
  const int* dst = edge + nE;

  // workspace: dinv[nN] | acc1[nN*8] | hw[nN*2]  (44 MB)
  float* dinv = (float*)d_ws;
  float* acc1 = dinv + nN;
  float* hw   = acc1 + (size_t)nN * 8;
  float* out  = (float*)d_out;

  const int B = 256;
  int gN    = (nN + B - 1) / B;
  int gE    = (nE + B - 1) / B;
  int waves = (nN + 15) / 16;                 // one wave32 per 16 nodes
  int gW    = ((waves * 32) + B - 1) / B;

  k_zero<<<gN, B, 0, stream>>>(dinv, nN);
  k_degree<<<gE, B, 0, stream>>>(dst, dinv, nE);
  k_node1<<<gW, B, 0, stream>>>(x, W1, dinv, acc1, nN);
  k_edge1<<<gE, B, 0, stream>>>(src, dst, x, W1, dinv, acc1, nE);
  k_node2<<<gW, B, 0, stream>>>(acc1, b1, W2, b2, dinv, hw, out, nN);
  k_edge2<<<gE, B, 0, stream>>>(src, dst, hw, dinv, out, nE);
}